// ADGraphModel_14826227105949
// MI455X (gfx1250) — compile-verified
//
#include <hip/hip_runtime.h>
#include <hip/hip_bf16.h>
#include <math.h>

typedef __attribute__((ext_vector_type(2))) float v2f;
typedef __attribute__((ext_vector_type(8))) float v8f;

#define N_NODES 100000
#define N_EDGES 1600000
#define IN_CH   64
#define HID     16
#define HEADS   4
#define NGRAPH  512
#define SLOPE   0.2f

// ---------------------------------------------------------------------------
// WMMA fp32 GEMM: Out[M x (NTILES*16)] = A[M x K] @ W[K x (NTILES*16)]
// One wave owns a 16-row strip. Per 4-wide k-step it loads one A fragment
// (v2f per lane) and feeds NTILES back-to-back V_WMMA_F32_16X16X4_F32 ops,
// with the B fragments served from LDS (W staged once per block).
//   A 16x4 : lanes 0-15 rows M=0..15 K={0,1}; lanes 16-31 K={2,3}
//   B 4x16 : VGPR0 = K=0 (lanes0-15)/K=2 (lanes16-31); VGPR1 = K=1/K=3
//   C/D    : VGPR i -> row i (lanes0-15) / row i+8 (lanes16-31)
// M must be a multiple of 16 (100000 = 6250*16).
// ---------------------------------------------------------------------------
template <int NTILES>
__global__ void wmma_gemm_f32(const float* __restrict__ A,
                              const float* __restrict__ W,
                              float* __restrict__ Out,
                              int K, int Mtiles) {
    constexpr int Ncol = NTILES * 16;
    extern __shared__ float Wlds[];                 // K * Ncol floats
    for (int i = threadIdx.x; i < K * Ncol; i += blockDim.x) Wlds[i] = W[i];
    __syncthreads();

    int lane  = threadIdx.x & 31;
    int wave  = threadIdx.x >> 5;
    int mtile = blockIdx.x * (blockDim.x >> 5) + wave;
    if (mtile >= Mtiles) return;                    // wave-uniform exit
    int m0 = mtile << 4;
    int hi = lane >> 4;                             // 0 or 1
    int lo = lane & 15;
    int arow = m0 + lo;

    v8f acc[NTILES];
#pragma unroll
    for (int nt = 0; nt < NTILES; ++nt) acc[nt] = (v8f){};

    for (int k = 0; k < K; k += 4) {
        int kk = k + hi * 2;
        v2f a;
        a.x = A[(size_t)arow * K + kk];
        a.y = A[(size_t)arow * K + kk + 1];
#pragma unroll
        for (int nt = 0; nt < NTILES; ++nt) {
            int bcol = nt * 16 + lo;
            v2f b;
            b.x = Wlds[kk * Ncol + bcol];
            b.y = Wlds[(kk + 1) * Ncol + bcol];
            acc[nt] = __builtin_amdgcn_wmma_f32_16x16x4_f32(
                /*neg_a=*/false, a, /*neg_b=*/false, b,
                /*c_mod=*/(short)0, acc[nt], /*reuse_a=*/false, /*reuse_b=*/false);
        }
    }

    int rbase = m0 + hi * 8;
#pragma unroll
    for (int nt = 0; nt < NTILES; ++nt) {
        int bcol = nt * 16 + lo;
#pragma unroll
        for (int i = 0; i < 8; ++i)
            Out[(size_t)(rbase + i) * Ncol + bcol] = acc[nt][i];
    }
}

// ---------------------------------------------------------------------------
// Per-(node, head) attention logits: alS = <h, a_src>, alD = <h, a_dst>
// ---------------------------------------------------------------------------
__global__ void att_logits(const float* __restrict__ H,
                           const float* __restrict__ a_src,
                           const float* __restrict__ a_dst,
                           float* __restrict__ alS, float* __restrict__ alD,
                           int n, int heads, int C) {
    int t = blockIdx.x * blockDim.x + threadIdx.x;
    if (t >= n * heads) return;
    int h = t % heads;
    const float* hp = H + (size_t)t * C;
    const float* as = a_src + h * C;
    const float* ad = a_dst + h * C;
    float s = 0.f, d = 0.f;
#pragma unroll 4
    for (int c = 0; c < C; ++c) { float v = hp[c]; s += v * as[c]; d += v * ad[c]; }
    alS[t] = s;
    alD[t] = d;
}

__global__ void fill_f32(float* __restrict__ p, float v, int n) {
    int i = blockIdx.x * blockDim.x + threadIdx.x;
    if (i < n) p[i] = v;
}

__device__ __forceinline__ void atomicMaxF(float* addr, float val) {
    // IEEE ordering trick: non-negative -> signed-int max, negative -> uint min
    if (val >= 0.f) atomicMax((int*)addr, __float_as_int(val));
    else            atomicMin((unsigned int*)addr, __float_as_uint(val));
}

__device__ __forceinline__ void edge_src_dst(const long long* __restrict__ ei,
                                             int e, long long& s, long long& d) {
    if (e < N_EDGES) { s = ei[e]; d = ei[N_EDGES + e]; }
    else             { s = d = (long long)(e - N_EDGES); }   // self-loop
}

// Pass A: segment max of leaky-relu logits over dst
__global__ void edge_max(const long long* __restrict__ ei, int total_edges, int heads,
                         const float* __restrict__ alS, const float* __restrict__ alD,
                         float* __restrict__ m) {
    int t = blockIdx.x * blockDim.x + threadIdx.x;
    if (t >= total_edges * heads) return;
    int e = t / heads, h = t % heads;
    long long s, d; edge_src_dst(ei, e, s, d);
    float v = alS[s * heads + h] + alD[d * heads + h];
    v = (v > 0.f) ? v : SLOPE * v;
    atomicMaxF(&m[d * heads + h], v);
}

// Pass B: denom = segment_sum(exp(e - m[dst]))
__global__ void edge_den(const long long* __restrict__ ei, int total_edges, int heads,
                         const float* __restrict__ alS, const float* __restrict__ alD,
                         const float* __restrict__ m, float* __restrict__ den) {
    int t = blockIdx.x * blockDim.x + threadIdx.x;
    if (t >= total_edges * heads) return;
    int e = t / heads, h = t % heads;
    long long s, d; edge_src_dst(ei, e, s, d);
    float v = alS[s * heads + h] + alD[d * heads + h];
    v = (v > 0.f) ? v : SLOPE * v;
    float p = expf(v - m[d * heads + h]);
    atomicAdd(&den[d * heads + h], p);
}

// Pass C: agg[dst] += h[src] * alpha
__global__ void edge_agg(const long long* __restrict__ ei, int total_edges, int heads, int C,
                         const float* __restrict__ alS, const float* __restrict__ alD,
                         const float* __restrict__ m, const float* __restrict__ den,
                         const float* __restrict__ H, float* __restrict__ agg) {
    int t = blockIdx.x * blockDim.x + threadIdx.x;
    if (t >= total_edges * heads) return;
    int e = t / heads, h = t % heads;
    long long s, d; edge_src_dst(ei, e, s, d);
    float v = alS[s * heads + h] + alD[d * heads + h];
    v = (v > 0.f) ? v : SLOPE * v;
    float p = expf(v - m[d * heads + h]);
    float alpha = p / (den[d * heads + h] + 1e-16f);
    const float* hp = H + ((size_t)s * heads + h) * C;
    float* op = agg + ((size_t)d * heads + h) * C;
#pragma unroll 4
    for (int c = 0; c < C; ++c) atomicAdd(&op[c], hp[c] * alpha);
}

// out = elu(agg + bias)
__global__ void bias_elu(const float* __restrict__ agg, const float* __restrict__ bias,
                         float* __restrict__ out, int n, int F) {
    int t = blockIdx.x * blockDim.x + threadIdx.x;
    if (t >= n * F) return;
    float v = agg[t] + bias[t % F];
    out[t] = (v > 0.f) ? v : (expf(v) - 1.f);
}

__global__ void pool_sum(const float* __restrict__ h, const long long* __restrict__ batch,
                         float* __restrict__ pooled, float* __restrict__ counts,
                         int n, int C) {
    int t = blockIdx.x * blockDim.x + threadIdx.x;
    if (t >= n) return;
    int g = (int)batch[t];
    atomicAdd(&counts[g], 1.f);
    const float* hp = h + (size_t)t * C;
    float* pp = pooled + (size_t)g * C;
#pragma unroll 4
    for (int c = 0; c < C; ++c) atomicAdd(&pp[c], hp[c]);
}

__global__ void head_kernel(const float* __restrict__ pooled, const float* __restrict__ counts,
                            const float* __restrict__ Wc, const float* __restrict__ bc,
                            float* __restrict__ out, int g_total, int C) {
    int g = blockIdx.x * blockDim.x + threadIdx.x;
    if (g >= g_total) return;
    float cnt = counts[g]; cnt = (cnt > 1.f) ? cnt : 1.f;
    float s = 0.f;
    for (int c = 0; c < C; ++c) s += (pooled[(size_t)g * C + c] / cnt) * Wc[c];
    out[g] = s + bc[0];
}

static inline unsigned gsz(long long n, int b) { return (unsigned)((n + b - 1) / b); }

extern "C" void kernel_launch(void* const* d_in, const int* in_sizes, int n_in,
                              void* d_out, int out_size, void* d_ws, size_t ws_size,
                              hipStream_t stream) {
    const float*     x      = (const float*)d_in[0];
    const long long* ei     = (const long long*)d_in[1];   // int64 [2,E]
    const long long* batch  = (const long long*)d_in[2];   // int64 [N]
    const float*     W1     = (const float*)d_in[3];
    const float*     a1_src = (const float*)d_in[4];
    const float*     a1_dst = (const float*)d_in[5];
    const float*     b1     = (const float*)d_in[6];
    const float*     W2     = (const float*)d_in[7];
    const float*     a2_src = (const float*)d_in[8];
    const float*     a2_dst = (const float*)d_in[9];
    const float*     b2     = (const float*)d_in[10];
    const float*     Wc     = (const float*)d_in[11];
    const float*     bc     = (const float*)d_in[12];
    float* out = (float*)d_out;

    const int Nn = N_NODES, ET = N_EDGES + N_NODES;
    const int F1 = HEADS * HID;          // 64

    // workspace carve-up (floats)
    float* ws = (float*)d_ws;
    size_t o = 0;
    float* h1   = ws + o; o += (size_t)Nn * F1;     // also holds elu(h1) later
    float* agg1 = ws + o; o += (size_t)Nn * F1;
    float* alS1 = ws + o; o += (size_t)Nn * HEADS;
    float* alD1 = ws + o; o += (size_t)Nn * HEADS;
    float* m1   = ws + o; o += (size_t)Nn * HEADS;
    float* den1 = ws + o; o += (size_t)Nn * HEADS;
    float* h2   = ws + o; o += (size_t)Nn * HID;    // also holds elu(h2) later
    float* agg2 = ws + o; o += (size_t)Nn * HID;
    float* alS2 = ws + o; o += (size_t)Nn;
    float* alD2 = ws + o; o += (size_t)Nn;
    float* m2   = ws + o; o += (size_t)Nn;
    float* den2 = ws + o; o += (size_t)Nn;
    float* pooled = ws + o; o += (size_t)NGRAPH * HID;
    float* counts = ws + o; o += (size_t)NGRAPH;

    const float NEG_INF = -__builtin_inff();
    const int Mtiles = Nn / 16;          // 6250, exact

    // ---- Layer 1 ----
    // h1 = x @ W1   (M=100000, K=64, Ncol=64), 4 waves/block, W1 staged in LDS
    wmma_gemm_f32<4><<<gsz(Mtiles, 4), 128, IN_CH * F1 * sizeof(float), stream>>>(
        x, W1, h1, IN_CH, Mtiles);
    att_logits<<<gsz((long long)Nn * HEADS, 256), 256, 0, stream>>>(
        h1, a1_src, a1_dst, alS1, alD1, Nn, HEADS, HID);
    fill_f32<<<gsz((long long)Nn * HEADS, 256), 256, 0, stream>>>(m1, NEG_INF, Nn * HEADS);
    fill_f32<<<gsz((long long)Nn * HEADS, 256), 256, 0, stream>>>(den1, 0.f, Nn * HEADS);
    fill_f32<<<gsz((long long)Nn * F1, 256), 256, 0, stream>>>(agg1, 0.f, Nn * F1);
    edge_max<<<gsz((long long)ET * HEADS, 256), 256, 0, stream>>>(ei, ET, HEADS, alS1, alD1, m1);
    edge_den<<<gsz((long long)ET * HEADS, 256), 256, 0, stream>>>(ei, ET, HEADS, alS1, alD1, m1, den1);
    edge_agg<<<gsz((long long)ET * HEADS, 256), 256, 0, stream>>>(
        ei, ET, HEADS, HID, alS1, alD1, m1, den1, h1, agg1);
    bias_elu<<<gsz((long long)Nn * F1, 256), 256, 0, stream>>>(agg1, b1, h1, Nn, F1);

    // ---- Layer 2 ----
    // h2 = elu(h1) @ W2   (M=100000, K=64, Ncol=16), W2 staged in LDS
    wmma_gemm_f32<1><<<gsz(Mtiles, 4), 128, F1 * HID * sizeof(float), stream>>>(
        h1, W2, h2, F1, Mtiles);
    att_logits<<<gsz((long long)Nn, 256), 256, 0, stream>>>(
        h2, a2_src, a2_dst, alS2, alD2, Nn, 1, HID);
    fill_f32<<<gsz((long long)Nn, 256), 256, 0, stream>>>(m2, NEG_INF, Nn);
    fill_f32<<<gsz((long long)Nn, 256), 256, 0, stream>>>(den2, 0.f, Nn);
    fill_f32<<<gsz((long long)Nn * HID, 256), 256, 0, stream>>>(agg2, 0.f, Nn * HID);
    edge_max<<<gsz((long long)ET, 256), 256, 0, stream>>>(ei, ET, 1, alS2, alD2, m2);
    edge_den<<<gsz((long long)ET, 256), 256, 0, stream>>>(ei, ET, 1, alS2, alD2, m2, den2);
    edge_agg<<<gsz((long long)ET, 256), 256, 0, stream>>>(
        ei, ET, 1, HID, alS2, alD2, m2, den2, h2, agg2);
    bias_elu<<<gsz((long long)Nn * HID, 256), 256, 0, stream>>>(agg2, b2, h2, Nn, HID);

    // ---- Pool + head ----
    fill_f32<<<gsz((long long)NGRAPH * (HID + 1), 256), 256, 0, stream>>>(
        pooled, 0.f, NGRAPH * (HID + 1));   // pooled and counts are contiguous
    pool_sum<<<gsz((long long)Nn, 256), 256, 0, stream>>>(h2, batch, pooled, counts, Nn, HID);
    head_kernel<<<gsz((long long)NGRAPH, 256), 256, 0, stream>>>(
        pooled, counts, Wc, bc, out, NGRAPH, HID);
}